// LFTSAD__3410204033106
// MI455X (gfx1250) — compile-verified
//
#include <hip/hip_runtime.h>

typedef __attribute__((ext_vector_type(2))) float v2f;
typedef __attribute__((ext_vector_type(8))) float v8f;

#define WINL 210
#define BATCH 8
#define MCH 55
#define BM (BATCH * MCH)   // 440
#define DM 256
#define SEQS 6
#define EPSF 1e-5f

// ---------------------------------------------------------------------------
// RevIN normalize: per (b,m) series of length WINL, write transposed (BM, L)
// ---------------------------------------------------------------------------
__global__ __launch_bounds__(128)
void revin_kernel(const float* __restrict__ x, float* __restrict__ xt) {
    const int bm = blockIdx.x;            // 0..439
    const int b = bm / MCH, m = bm % MCH;
    __shared__ float ssum[128];
    __shared__ float ssq[128];
    float s = 0.f, sq = 0.f;
    for (int l = threadIdx.x; l < WINL; l += 128) {
        float v = x[(b * WINL + l) * MCH + m];
        s += v; sq += v * v;
    }
    ssum[threadIdx.x] = s; ssq[threadIdx.x] = sq;
    __syncthreads();
    for (int off = 64; off > 0; off >>= 1) {
        if (threadIdx.x < off) {
            ssum[threadIdx.x] += ssum[threadIdx.x + off];
            ssq[threadIdx.x]  += ssq[threadIdx.x + off];
        }
        __syncthreads();
    }
    const float mean = ssum[0] * (1.0f / WINL);
    const float var  = ssq[0] * (1.0f / WINL) - mean * mean;   // ddof=0
    const float inv  = 1.0f / sqrtf(var + EPSF);
    for (int l = threadIdx.x; l < WINL; l += 128) {
        float v = x[(b * WINL + l) * MCH + m];
        xt[bm * WINL + l] = (v - mean) * inv;
    }
}

// ---------------------------------------------------------------------------
// Fused fc1(WMMA f32 16x16x4) -> ReLU -> fc2 for one branch.
// TYPE: 0=size, 1=num, 2=size_seq, 3=num_seq.  Block = 128 thr (4 waves),
// tile = 16 rows x 256 hidden; activations live in LDS only.
// Each wave owns 4 hidden N-tiles, processed as 2 pairs with dual
// accumulator chains sharing one A fragment per K-step.
// ---------------------------------------------------------------------------
template <int TYPE, int P, int N, int F, int O>
__global__ __launch_bounds__(128)
void branch_kernel(const float* __restrict__ xt,
                   const float* __restrict__ W1, const float* __restrict__ b1,
                   const float* __restrict__ W2, const float* __restrict__ b2,
                   float* __restrict__ out, int q)
{
    constexpr int F4      = (F + 3) & ~3;      // K padded to multiple of 4
    constexpr int KMAIN   = (F / 4) * 4;       // unconditional K range
    constexpr int rowsPer = (TYPE <= 1) ? WINL : (P * N);
    constexpr int R       = BM * rowsPer;

    const int tid   = threadIdx.x;
    const int wave  = tid >> 5;
    const int lane  = tid & 31;
    const int row0  = blockIdx.x * 16;

    __shared__ float fbuf[16][F4 + 1];      // gathered features (zero padded)
    __shared__ float hbuf[16][DM + 1];      // ReLU(fc1) activations

    if (tid == 0) {
        __builtin_prefetch(W1, 0, 1);       // global_prefetch_b8 (L2-resident)
        __builtin_prefetch(W2, 0, 1);
    }

    // ---- gather leave-one-out feature tile into LDS -----------------------
    for (int idx = tid; idx < 16 * F4; idx += 128) {
        const int rr = idx / F4, f = idx % F4;
        const int rho = row0 + rr;
        float val = 0.f;
        if (rho < R && f < F) {
            const int bm = rho / rowsPer;
            const int j  = rho % rowsPer;
            int src;
            if (TYPE == 0) {                 // size: loo over patch axis
                const int i = j / N, k = j % N;
                src = (f + (f >= i)) * N + k;
            } else if (TYPE == 1) {          // num: loo over position axis
                const int r = j / N, k = j % N;
                src = r * N + (f + (f >= k));
            } else if (TYPE == 2) {          // size_seq
                const int r = j / N, k = j % N;
                const int i2 = f / SEQS, t = f % SEQS;
                src = ((i2 + (i2 >= r)) * N + k) * SEQS + t;
            } else {                         // num_seq (row order j = k*P + r)
                const int k = j / P, r = j % P;
                const int k2 = f / SEQS, t = f % SEQS;
                src = (r * N + (k2 + (k2 >= k))) * SEQS + t;
            }
            val = xt[bm * WINL + src];
        }
        fbuf[rr][f] = val;
    }
    __syncthreads();

    // ---- fc1: f32 WMMA 16x16x4, dual accumulators per wave ----------------
    const int halfSel = lane >> 4;           // 0: lanes 0-15, 1: lanes 16-31
    const int lane16  = lane & 15;
    #pragma unroll
    for (int pp = 0; pp < 2; ++pp) {
        const int nt0 = wave * 4 + pp * 2;   // this wave's tile pair
        const int nt1 = nt0 + 1;
        const int h0 = nt0 * 16 + lane16;    // hidden channel (B fragment N)
        const int h1 = nt1 * 16 + lane16;
        const float bias0 = b1[h0];
        const float bias1 = b1[h1];
        v8f c0, c1;
        #pragma unroll
        for (int v = 0; v < 8; ++v) { c0[v] = bias0; c1[v] = bias1; }

        // main K range: unconditional loads, fully unrolled
        #pragma unroll
        for (int kk = 0; kk < KMAIN; kk += 4) {
            const int kA = kk + 2 * halfSel; // lanes 0-15: K={0,1}; 16-31: K={2,3}
            v2f a, b0, b1v;
            a.x   = fbuf[lane16][kA];
            a.y   = fbuf[lane16][kA + 1];
            b0.x  = W1[h0 * F + kA];
            b0.y  = W1[h0 * F + kA + 1];
            b1v.x = W1[h1 * F + kA];
            b1v.y = W1[h1 * F + kA + 1];
            c0 = __builtin_amdgcn_wmma_f32_16x16x4_f32(
                     false, a, false, b0, (short)0, c0, false, false);
            c1 = __builtin_amdgcn_wmma_f32_16x16x4_f32(
                     false, a, false, b1v, (short)0, c1, false, false);
        }
        // tail K step (only when F % 4 != 0): branch-free clamp + zero-select
        if constexpr (KMAIN < F4) {
            const int kA  = KMAIN + 2 * halfSel;
            const int k0c = (kA     < F) ? kA     : 0;
            const int k1c = (kA + 1 < F) ? kA + 1 : 0;
            const float m0 = (kA     < F) ? 1.f : 0.f;
            const float m1 = (kA + 1 < F) ? 1.f : 0.f;
            v2f a, b0, b1v;
            a.x   = fbuf[lane16][kA];        // fbuf zero-padded, no guard
            a.y   = fbuf[lane16][kA + 1];
            b0.x  = W1[h0 * F + k0c] * m0;
            b0.y  = W1[h0 * F + k1c] * m1;
            b1v.x = W1[h1 * F + k0c] * m0;
            b1v.y = W1[h1 * F + k1c] * m1;
            c0 = __builtin_amdgcn_wmma_f32_16x16x4_f32(
                     false, a, false, b0, (short)0, c0, false, false);
            c1 = __builtin_amdgcn_wmma_f32_16x16x4_f32(
                     false, a, false, b1v, (short)0, c1, false, false);
        }

        // ReLU + stash activations (C/D: VGPR v -> row v, +8 for hi lanes)
        #pragma unroll
        for (int v = 0; v < 8; ++v) {
            float v0 = c0[v] > 0.f ? c0[v] : 0.f;
            float v1 = c1[v] > 0.f ? c1[v] : 0.f;
            const int rr = v + 8 * halfSel;
            hbuf[rr][h0] = v0;
            hbuf[rr][h1] = v1;
        }
    }
    __syncthreads();

    // ---- fc2 + branch-specific output scatter -----------------------------
    if (tid < 16 * O) {
        const int rr = tid / O, o = tid % O;
        const int rho = row0 + rr;
        if (rho < R) {
            float acc = b2[o];
            #pragma unroll 8
            for (int h = 0; h < DM; ++h) acc += hbuf[rr][h] * W2[o * DM + h];

            const int bm = rho / rowsPer;
            const int j  = rho % rowsPer;
            int ob = bm / MCH, om = bm % MCH, l;
            if (TYPE == 0) {
                l = j;
            } else if (TYPE == 1) {
                // faithful (n,p,BM)-transpose reshape scramble
                const int r = j / N, k = j % N;
                const int flat = (k * P + r) * BM + bm;
                ob = flat / (MCH * WINL);
                const int rem = flat % (MCH * WINL);
                om = rem / WINL;
                l  = rem % WINL;
            } else if (TYPE == 2) {
                l = j * SEQS + o;
            } else {
                // num_seq regroup with hard-coded d0 = 5
                const int cgrp = j / 5, dd = j % 5;
                l = (dd * ((N * P) / 5) + cgrp) * SEQS + o;
            }
            out[((q * BATCH + ob) * WINL + l) * MCH + om] = acc;
        }
    }
}

// ---------------------------------------------------------------------------
extern "C" void kernel_launch(void* const* d_in, const int* in_sizes, int n_in,
                              void* d_out, int out_size, void* d_ws, size_t ws_size,
                              hipStream_t stream) {
    (void)in_sizes; (void)n_in; (void)out_size; (void)ws_size;
    const float* x  = (const float*)d_in[0];
    float*       xt = (float*)d_ws;             // 440*210*4 = 369,600 B
    float*       out = (float*)d_out;

    revin_kernel<<<BM, 128, 0, stream>>>(x, xt);

#define LAUNCH_BRANCH(TYPE, P, N, F, O, Q, WI)                                   \
    {                                                                            \
        constexpr int rp = ((TYPE) <= 1) ? WINL : ((P) * (N));                   \
        constexpr int R  = BM * rp;                                              \
        branch_kernel<TYPE, P, N, F, O><<<(R + 15) / 16, 128, 0, stream>>>(      \
            xt, (const float*)d_in[WI], (const float*)d_in[(WI) + 1],            \
            (const float*)d_in[(WI) + 2], (const float*)d_in[(WI) + 3], out, Q); \
    }

    // size branches (s1_j/s2_j): q = 0..2
    LAUNCH_BRANCH(0, 3, 70, 2, 1, 0, 1);
    LAUNCH_BRANCH(0, 5, 42, 4, 1, 1, 9);
    LAUNCH_BRANCH(0, 7, 30, 6, 1, 2, 17);
    // num branches (n1_j/n2_j): q = 3..5
    LAUNCH_BRANCH(1, 3, 70, 69, 1, 3, 5);
    LAUNCH_BRANCH(1, 5, 42, 41, 1, 4, 13);
    LAUNCH_BRANCH(1, 7, 30, 29, 1, 5, 21);
    // size_seq branches (ss1_j/ss2_j): q = 6..7
    LAUNCH_BRANCH(2, 5, 7, 24, 6, 6, 25);
    LAUNCH_BRANCH(2, 7, 5, 36, 6, 7, 33);
    // num_seq branches (ns1_j/ns2_j): q = 8..9
    LAUNCH_BRANCH(3, 5, 7, 36, 6, 8, 29);
    LAUNCH_BRANCH(3, 7, 5, 24, 6, 9, 37);
#undef LAUNCH_BRANCH
}